// NeRFRenderer_13589276525205
// MI455X (gfx1250) — compile-verified
//
#include <hip/hip_runtime.h>
#include <hip/hip_bf16.h>

// NeRF ray-marching renderer for gfx1250 (MI455X).
//
// Roofline: ~2.5 GFLOP total (irrelevant) vs ~33.5M trilerp samples gathering
// 8 x 16B corners each. Grids (32 MB packed) are resident in the 192 MB L2,
// so the kernel is bound by L2 line traffic + gather latency. WMMA/TDM do not
// apply (per-lane unique gathers, no coherent tiles), so we optimize data
// movement:
//   * float4 {r,g,b,sigma} AoS  -> one b128 load per corner
//   * 2x2x2 cacheline blocking  -> 8 cells = exactly one 128B line;
//     expected lines/sample drops from ~5 to ~3.4 ((1.5)^3), all aligned
//   * early ray termination, gfx1250 global_prefetch_b8 for next step's lines
//   * wave32-friendly 256-thread blocks (8 waves) -> 8192 waves of latency hiding

#define GRES   128
#define NSTEPS 128

__device__ __forceinline__ float4 lerp4(const float4 a, const float4 b, const float t) {
    return make_float4(fmaf(t, b.x - a.x, a.x),
                       fmaf(t, b.y - a.y, a.y),
                       fmaf(t, b.z - a.z, a.z),
                       fmaf(t, b.w - a.w, a.w));
}

// Pack sigma (G^3) + rgb (G^3 x 3) into float4 {r,g,b,sigma} per voxel,
// scattered into a 2x2x2-blocked layout: block (x>>1,y>>1,z>>1) of 64^3 blocks,
// flat block index (bx<<12)+(by<<6)+bz, 8 cells per block (one 128B line),
// local offset (x&1)<<2 | (y&1)<<1 | (z&1).
__global__ __launch_bounds__(256) void nerf_repack(
    const float* __restrict__ sigma, const float* __restrict__ rgb,
    float4* __restrict__ packed, int n)
{
    int i = blockIdx.x * 256 + threadIdx.x;
    if (i >= n) return;
    const int x = i >> 14, y = (i >> 7) & 127, z = i & 127;
    const int dst = ((((x >> 1) << 12) + ((y >> 1) << 6) + (z >> 1)) << 3)
                  + ((x & 1) << 2) + ((y & 1) << 1) + (z & 1);
    packed[dst] = make_float4(rgb[3 * i + 0], rgb[3 * i + 1], rgb[3 * i + 2], sigma[i]);
}

template <bool PACKED>
__global__ __launch_bounds__(256) void nerf_render(
    const float* __restrict__ rays_o, const float* __restrict__ rays_d,
    const float4* __restrict__ grid4,
    const float* __restrict__ sigma_g, const float* __restrict__ rgb_g,
    const float* __restrict__ bg, float* __restrict__ out, int nrays)
{
    int r = blockIdx.x * 256 + threadIdx.x;
    if (r >= nrays) return;

    const float ox = rays_o[3 * r + 0], oy = rays_o[3 * r + 1], oz = rays_o[3 * r + 2];
    const float dx = rays_d[3 * r + 0], dy = rays_d[3 * r + 1], dz = rays_d[3 * r + 2];

    // Slab test vs AABB [-1,-0.5,-1]..[1,0.5,1], matching the reference.
    const float sdx = (fabsf(dx) < 1e-9f) ? 1e-9f : dx;
    const float sdy = (fabsf(dy) < 1e-9f) ? 1e-9f : dy;
    const float sdz = (fabsf(dz) < 1e-9f) ? 1e-9f : dz;
    const float t1x = (-1.0f - ox) / sdx, t2x = (1.0f - ox) / sdx;
    const float t1y = (-0.5f - oy) / sdy, t2y = (0.5f - oy) / sdy;
    const float t1z = (-1.0f - oz) / sdz, t2z = (1.0f - oz) / sdz;
    float nearT = fmaxf(fmaxf(fminf(t1x, t2x), fminf(t1y, t2y)), fminf(t1z, t2z));
    float farT  = fminf(fminf(fmaxf(t1x, t2x), fmaxf(t1y, t2y)), fmaxf(t1z, t2z));
    nearT = fmaxf(nearT, 0.05f);
    farT  = fmaxf(farT, nearT + 1e-6f);
    const float dt = (farT - nearT) * (1.0f / (float)NSTEPS);

    // Per-step motion in voxel space (for next-step prefetch only).
    const float vdx = dt * dx * 63.5f;
    const float vdy = dt * dy * 127.0f;
    const float vdz = dt * dz * 63.5f;

    float T = 1.0f, ws = 0.0f, ir = 0.0f, ig = 0.0f, ib = 0.0f;

    for (int i = 0; i < NSTEPS; ++i) {
        if (T <= 1e-4f) break;  // exact equivalent of reference alpha masking

        const float t  = fmaf((float)i + 0.5f, dt, nearT);
        const float px = fmaf(t, dx, ox);
        const float py = fmaf(t, dy, oy);
        const float pz = fmaf(t, dz, oz);

        // World -> grid coords in [0, G-1].
        float ux = fminf(fmaxf((px + 1.0f) * 63.5f,  0.0f), 127.0f);
        float uy = fminf(fmaxf((py + 0.5f) * 127.0f, 0.0f), 127.0f);
        float uz = fminf(fmaxf((pz + 1.0f) * 63.5f,  0.0f), 127.0f);
        const float fx0 = fminf(floorf(ux), 126.0f);
        const float fy0 = fminf(floorf(uy), 126.0f);
        const float fz0 = fminf(floorf(uz), 126.0f);
        const float fx = ux - fx0, fy = uy - fy0, fz = uz - fz0;
        const int ix = (int)fx0, iy = (int)fy0, iz = (int)fz0;

        float4 c000, c001, c010, c011, c100, c101, c110, c111;
        if (PACKED) {
            // Blocked addressing: base cell + parity-dependent axis strides.
            const int blk  = ((ix >> 1) << 12) + ((iy >> 1) << 6) + (iz >> 1);
            const int base = (blk << 3) + ((ix & 1) << 2) + ((iy & 1) << 1) + (iz & 1);
            const int dxo  = (ix & 1) ? (32768 - 4) : 4;  // +x: next plane-block or in-block
            const int dyo  = (iy & 1) ? (512 - 2)   : 2;  // +y: next row-block or in-block
            const int dzo  = (iz & 1) ? (8 - 1)     : 1;  // +z: next block or in-block

            // gfx1250: prefetch next step's base block line + its +x block line.
            {
                const float nx = fminf(fmaxf(ux + vdx, 0.0f), 126.0f);
                const float ny = fminf(fmaxf(uy + vdy, 0.0f), 126.0f);
                const float nz = fminf(fmaxf(uz + vdz, 0.0f), 126.0f);
                const int bxn = ((int)nx) >> 1;
                const int bn  = (bxn << 12) + ((((int)ny) >> 1) << 6) + (((int)nz) >> 1);
                const int bn2 = bn + ((bxn < 63) ? 4096 : 0);
                __builtin_prefetch((const void*)(grid4 + (bn  << 3)), 0, 0);
                __builtin_prefetch((const void*)(grid4 + (bn2 << 3)), 0, 0);
            }

            const float4* p = grid4 + base;
            c000 = p[0];               c001 = p[dzo];
            c010 = p[dyo];             c011 = p[dyo + dzo];
            c100 = p[dxo];             c101 = p[dxo + dzo];
            c110 = p[dxo + dyo];       c111 = p[dxo + dyo + dzo];
        } else {
            // Fallback: gather directly from the two flat grids.
            const int base = (ix << 14) + (iy << 7) + iz;
            {
                const float nx = fminf(fmaxf(ux + vdx, 0.0f), 126.0f);
                const float ny = fminf(fmaxf(uy + vdy, 0.0f), 126.0f);
                const float nz = fminf(fmaxf(uz + vdz, 0.0f), 126.0f);
                const int pb = (((int)nx) << 14) + (((int)ny) << 7) + ((int)nz);
                __builtin_prefetch((const void*)(sigma_g + pb), 0, 0);
                __builtin_prefetch((const void*)(rgb_g + 3 * pb), 0, 0);
            }
            const int o001 = base + 1;
            const int o010 = base + GRES,        o011 = base + GRES + 1;
            const int o100 = base + GRES * GRES, o101 = base + GRES * GRES + 1;
            const int o110 = o100 + GRES,        o111 = o100 + GRES + 1;
            c000 = make_float4(rgb_g[3*base+0], rgb_g[3*base+1], rgb_g[3*base+2], sigma_g[base]);
            c001 = make_float4(rgb_g[3*o001+0], rgb_g[3*o001+1], rgb_g[3*o001+2], sigma_g[o001]);
            c010 = make_float4(rgb_g[3*o010+0], rgb_g[3*o010+1], rgb_g[3*o010+2], sigma_g[o010]);
            c011 = make_float4(rgb_g[3*o011+0], rgb_g[3*o011+1], rgb_g[3*o011+2], sigma_g[o011]);
            c100 = make_float4(rgb_g[3*o100+0], rgb_g[3*o100+1], rgb_g[3*o100+2], sigma_g[o100]);
            c101 = make_float4(rgb_g[3*o101+0], rgb_g[3*o101+1], rgb_g[3*o101+2], sigma_g[o101]);
            c110 = make_float4(rgb_g[3*o110+0], rgb_g[3*o110+1], rgb_g[3*o110+2], sigma_g[o110]);
            c111 = make_float4(rgb_g[3*o111+0], rgb_g[3*o111+1], rgb_g[3*o111+2], sigma_g[o111]);
        }

        // Trilinear lerp chain (algebraically identical to the 8-weight sum).
        const float4 a  = lerp4(lerp4(c000, c001, fz), lerp4(c010, c011, fz), fy);
        const float4 b  = lerp4(lerp4(c100, c101, fz), lerp4(c110, c111, fz), fy);
        const float4 cv = lerp4(a, b, fx);

        const float sig   = (cv.w > 0.01f) ? cv.w : 0.0f;
        const float alpha = 1.0f - __expf(-sig * dt);
        const float w     = alpha * T;
        ws += w;
        ir = fmaf(w, cv.x, ir);
        ig = fmaf(w, cv.y, ig);
        ib = fmaf(w, cv.z, ib);
        T *= (1.0f - alpha);
    }

    const float rem = 1.0f - ws;
    const float b0 = bg[0], b1 = bg[1], b2 = bg[2];  // uniform -> s_loads
    out[3 * r + 0] = fminf(fmaxf(fmaf(rem, b0, ir), 0.0f), 1.0f);
    out[3 * r + 1] = fminf(fmaxf(fmaf(rem, b1, ig), 0.0f), 1.0f);
    out[3 * r + 2] = fminf(fmaxf(fmaf(rem, b2, ib), 0.0f), 1.0f);
}

extern "C" void kernel_launch(void* const* d_in, const int* in_sizes, int n_in,
                              void* d_out, int out_size, void* d_ws, size_t ws_size,
                              hipStream_t stream) {
    const float* rays_o = (const float*)d_in[0];
    const float* rays_d = (const float*)d_in[1];
    const float* sigma  = (const float*)d_in[2];
    const float* rgb    = (const float*)d_in[3];
    const float* bg     = (const float*)d_in[4];
    float* out = (float*)d_out;

    const int nrays   = in_sizes[0] / 3;      // 262144
    const int ncells  = in_sizes[2];          // 128^3
    const int rblocks = (nrays + 255) / 256;

    if (ws_size >= (size_t)ncells * sizeof(float4)) {
        float4* g4 = (float4*)d_ws;
        nerf_repack<<<(ncells + 255) / 256, 256, 0, stream>>>(sigma, rgb, g4, ncells);
        nerf_render<true><<<rblocks, 256, 0, stream>>>(rays_o, rays_d, g4,
                                                       sigma, rgb, bg, out, nrays);
    } else {
        nerf_render<false><<<rblocks, 256, 0, stream>>>(rays_o, rays_d, nullptr,
                                                        sigma, rgb, bg, out, nrays);
    }
}